// Hgru3_79182017069781
// MI455X (gfx1250) — compile-verified
//
#include <hip/hip_runtime.h>
#include <math.h>

#define EMBED 1024
#define DHEAD 128
#define NHEAD 8
#define BATCH 4
#define SEQT  2048
#define BT    (BATCH*SEQT)
#define PI_F  3.14159265358979f
#define CHUNK 16
#define SCALEQ 0.0625f   /* (2d)^-0.5 = 256^-0.5 */

typedef __attribute__((ext_vector_type(2))) float v2f;
typedef __attribute__((ext_vector_type(8))) float v8f;
typedef __attribute__((ext_vector_type(4))) int   v4i;

#define AS1 __attribute__((address_space(1)))
#define AS3 __attribute__((address_space(3)))

#if defined(__has_builtin)
#  if __has_builtin(__builtin_amdgcn_global_load_async_to_lds_b128)
#    define HAVE_ASYNC_LDS 1
#  endif
#  if __has_builtin(__builtin_amdgcn_s_wait_asynccnt)
#    define HAVE_WAIT_ASYNC 1
#  endif
#endif

// 16-byte memory->LDS copy: async DMA on gfx1250 toolchains that expose it,
// synchronous float4 copy otherwise.
__device__ __forceinline__ void cp16(const void* g, void* l) {
#if defined(HAVE_ASYNC_LDS)
  __builtin_amdgcn_global_load_async_to_lds_b128((AS1 v4i*)g, (AS3 v4i*)l,
                                                 0, 0);
#else
  *(float4*)l = *(const float4*)g;
#endif
}

__device__ __forceinline__ void wait_async() {
#if defined(HAVE_ASYNC_LDS)
#  if defined(HAVE_WAIT_ASYNC)
  __builtin_amdgcn_s_wait_asynccnt(0);
#  else
  asm volatile("s_wait_asynccnt 0" ::: "memory");
#  endif
#endif
}

__device__ __forceinline__ v8f wmma4(v2f a, v2f b, v8f c) {
  // V_WMMA_F32_16X16X4_F32 : (neg_a, A, neg_b, B, c_mod, C, reuse_a, reuse_b)
  return __builtin_amdgcn_wmma_f32_16x16x4_f32(false, a, false, b, (short)0, c,
                                               false, false);
}

__device__ __forceinline__ float sigmoidf_(float x) {
  return 1.0f / (1.0f + __expf(-x));
}

// ---------------------------------------------------------------------------
// fp32 WMMA GEMM: C = act(A[MxK] @ B[KxN]), row-major. M%128==0, N%64==0,
// K%32==0. act: 0=none, 1=silu, 2=sigmoid.
// 128x64 block tile, 256 threads = 8 waves, each wave owns a 2x2 grid of
// 16x16 accumulators (1 LDS fetch per WMMA). K-panels (32) are double-
// buffered and staged with async global->LDS DMA.
// ---------------------------------------------------------------------------
#define AKLD 36
#define BKLD 68
__global__ __launch_bounds__(256)
void gemm_wmma_f32(const float* __restrict__ A, const float* __restrict__ Bm,
                   float* __restrict__ C, int M, int N, int K, int act) {
  __shared__ __align__(16) float As[2][128 * AKLD];
  __shared__ __align__(16) float Bs[2][32 * BKLD];
  const int tid = threadIdx.x;
  const int w = tid >> 5, lane = tid & 31;
  const int h2 = lane >> 4, l15 = lane & 15;
  const int m2 = (w & 3) * 2, n2 = (w >> 2) * 2;
  const int bm = blockIdx.x * 128, bn = blockIdx.y * 64;
  v8f acc[4] = {};

  auto stage = [&](int buf, int k0) {
#pragma unroll
    for (int r = 0; r < 4; ++r) {          // As: 128x32 = 1024 float4
      int idx = tid + r * 256;
      int ar = idx >> 3, ac = (idx & 7) * 4;
      cp16(&A[(size_t)(bm + ar) * K + k0 + ac], &As[buf][ar * AKLD + ac]);
    }
#pragma unroll
    for (int r = 0; r < 2; ++r) {          // Bs: 32x64 = 512 float4
      int idx = tid + r * 256;
      int br = idx >> 4, bc = (idx & 15) * 4;
      cp16(&Bm[(size_t)(k0 + br) * N + bn + bc], &Bs[buf][br * BKLD + bc]);
    }
  };

  stage(0, 0);
  wait_async();
  __syncthreads();
  const int nk = K / 32;
  for (int p = 0; p < nk; ++p) {
    const int cur = p & 1;
    if (p + 1 < nk) stage(1 - cur, (p + 1) * 32);   // prefetch next panel
    const float* as = As[cur];
    const float* bs = Bs[cur];
#pragma unroll
    for (int kk = 0; kk < 32; kk += 4) {
      v2f a0, a1, b0, b1;
      a0.x = as[((m2 + 0) * 16 + l15) * AKLD + kk + 2 * h2];
      a0.y = as[((m2 + 0) * 16 + l15) * AKLD + kk + 2 * h2 + 1];
      a1.x = as[((m2 + 1) * 16 + l15) * AKLD + kk + 2 * h2];
      a1.y = as[((m2 + 1) * 16 + l15) * AKLD + kk + 2 * h2 + 1];
      b0.x = bs[(kk + 2 * h2) * BKLD + (n2 + 0) * 16 + l15];
      b0.y = bs[(kk + 2 * h2 + 1) * BKLD + (n2 + 0) * 16 + l15];
      b1.x = bs[(kk + 2 * h2) * BKLD + (n2 + 1) * 16 + l15];
      b1.y = bs[(kk + 2 * h2 + 1) * BKLD + (n2 + 1) * 16 + l15];
      acc[0] = wmma4(a0, b0, acc[0]);
      acc[1] = wmma4(a0, b1, acc[1]);
      acc[2] = wmma4(a1, b0, acc[2]);
      acc[3] = wmma4(a1, b1, acc[3]);
    }
    wait_async();
    __syncthreads();
  }
#pragma unroll
  for (int mi = 0; mi < 2; ++mi)
#pragma unroll
    for (int ni = 0; ni < 2; ++ni)
#pragma unroll
      for (int r = 0; r < 8; ++r) {
        int row = bm + (m2 + mi) * 16 + r + 8 * h2;
        int col = bn + (n2 + ni) * 16 + l15;
        float x0 = acc[mi * 2 + ni][r];
        if (act == 1) x0 = x0 * sigmoidf_(x0);
        else if (act == 2) x0 = sigmoidf_(x0);
        C[(size_t)row * N + col] = x0;
      }
}

// ---------------------------------------------------------------------------
// theta = cumsum_t(sign(f-0.5))*pi per (b, e) channel; emit cos/sin.
// ---------------------------------------------------------------------------
__global__ __launch_bounds__(256)
void cumsum_trig(const float* __restrict__ Kk, float* __restrict__ Cosb,
                 float* __restrict__ Sinb) {
  int g = blockIdx.x * 256 + threadIdx.x;   // 0..B*EMBED-1
  int b = g >> 10, e = g & 1023;
  size_t base = ((size_t)b * SEQT) * EMBED + e;
  int n = 0;
  for (int t = 0; t < SEQT; ++t) {
    size_t idx = base + (size_t)t * EMBED;
    float f = 1.f - Kk[idx];
    float d = f - 0.5f;
    n += (d > 0.f) - (d < 0.f);
    float th = (float)n * PI_F;
    Cosb[idx] = cosf(th);
    Sinb[idx] = sinf(th);
  }
}

// ---------------------------------------------------------------------------
// Chunked GLA scan; all matmuls via V_WMMA_F32_16X16X4_F32.
// Block = (jb, h, b): state S[256 x 64] in LDS. Raw chunk inputs
// (k,q,cos,sin,v) are double-buffered and prefetched with async DMA while
// the current chunk's WMMA phases run.
// Per 16-step chunk:  Qp = q*trig*Ainc*scale,  Kp = k*trig/Ainc,
//   O = Qp@S + tril(Qp@Kp^T)@V ,  S = diag(Afin)*(S + Kp^T@V).
// ---------------------------------------------------------------------------
#define R_K 0
#define R_Q 2048
#define R_C 4096
#define R_S 6144
#define R_V 8192      /* stride 72 per row */
#define RAWSZ 9344    /* 8192 + 16*72 */
__global__ __launch_bounds__(256)
void hgru3_scan(const float* __restrict__ Q, const float* __restrict__ Kk,
                const float* __restrict__ V, const float* __restrict__ Cosb,
                const float* __restrict__ Sinb, float* __restrict__ Oat) {
  __shared__ __align__(16) float S[256 * 66];
  __shared__ __align__(16) float Qp[16 * 257];
  __shared__ __align__(16) float Kp[16 * 257];
  __shared__ __align__(16) float Ainc[16 * 132];
  __shared__ __align__(16) float Afin[128];
  __shared__ __align__(16) float Pm[16 * 20];
  __shared__ __align__(16) float Raw[2][RAWSZ];
  const int jb = blockIdx.x;   // 0..1 : Dv half
  const int h  = blockIdx.y;   // 0..7
  const int b  = blockIdx.z;   // 0..3
  const int tid = threadIdx.x;
  const int w = tid >> 5, lane = tid & 31;
  const int h2 = lane >> 4, l15 = lane & 15;

  for (int e = tid; e < 256 * 66; e += 256) S[e] = 0.f;

  const size_t baseBH = ((size_t)b * SEQT) * EMBED + h * DHEAD;

  auto stage = [&](float* raw, int t0) {
#pragma unroll
    for (int r = 0; r < 2; ++r) {        // 16x128 arrays: 512 float4 each
      int idx = tid + r * 256;
      int t = idx >> 5, c = (idx & 31) * 4;
      size_t g = baseBH + (size_t)(t0 + t) * EMBED + c;
      cp16(&Kk[g],   &raw[R_K + t * 128 + c]);
      cp16(&Q[g],    &raw[R_Q + t * 128 + c]);
      cp16(&Cosb[g], &raw[R_C + t * 128 + c]);
      cp16(&Sinb[g], &raw[R_S + t * 128 + c]);
    }
    {                                    // v: 16x64 = 256 float4
      int t = tid >> 4, c = (tid & 15) * 4;
      cp16(&V[baseBH + (size_t)(t0 + t) * EMBED + jb * 64 + c],
           &raw[R_V + t * 72 + c]);
    }
  };

  stage(Raw[0], 0);
  wait_async();
  __syncthreads();

  for (int nc = 0; nc < SEQT / CHUNK; ++nc) {
    const int t0 = nc * CHUNK;
    float* raw  = Raw[nc & 1];
    if (nc + 1 < SEQT / CHUNK) stage(Raw[(nc + 1) & 1], t0 + CHUNK);
    const float* rv = raw + R_V;

    // --- A1: in-chunk cumulative decay (f = 1-k, duplicated halves) ---
    if (tid < 128) {
      float a = 1.f;
#pragma unroll
      for (int t = 0; t < 16; ++t) {
        a *= (1.f - raw[R_K + t * 128 + tid]);
        Ainc[t * 132 + tid] = a;
      }
      Afin[tid] = a;
    }
    __syncthreads();
    // --- A2: transformed operands ---
    for (int e = tid; e < 16 * 256; e += 256) {
      int t = e >> 8, i = e & 255, ii = i & 127;
      float trig = (i < 128) ? raw[R_C + t * 128 + ii] : raw[R_S + t * 128 + ii];
      float a = Ainc[t * 132 + ii];
      Qp[t * 257 + i] = raw[R_Q + t * 128 + ii] * trig * a * SCALEQ;
      Kp[t * 257 + i] = raw[R_K + t * 128 + ii] * trig / a;
    }
    __syncthreads();
    // --- B: O_inter = Qp@S (waves 0-3); P = Qp@Kp^T (wave 4) ---
    v8f acc = {};
    if (w < 4) {
      for (int k0 = 0; k0 < 256; k0 += 4) {
        v2f a;
        a.x = Qp[l15 * 257 + k0 + 2 * h2];
        a.y = Qp[l15 * 257 + k0 + 2 * h2 + 1];
        v2f bb;
        bb.x = S[(k0 + 2 * h2) * 66 + w * 16 + l15];
        bb.y = S[(k0 + 2 * h2 + 1) * 66 + w * 16 + l15];
        acc = wmma4(a, bb, acc);
      }
    } else if (w == 4) {
      v8f pacc = {};
      for (int k0 = 0; k0 < 256; k0 += 4) {
        v2f a;
        a.x = Qp[l15 * 257 + k0 + 2 * h2];
        a.y = Qp[l15 * 257 + k0 + 2 * h2 + 1];
        v2f bb;
        bb.x = Kp[l15 * 257 + k0 + 2 * h2];
        bb.y = Kp[l15 * 257 + k0 + 2 * h2 + 1];
        pacc = wmma4(a, bb, pacc);
      }
#pragma unroll
      for (int r = 0; r < 8; ++r) {
        int t = r + 8 * h2, s = l15;
        Pm[t * 20 + s] = (s <= t) ? pacc[r] : 0.f;
      }
    }
    __syncthreads();
    // --- C: U = Kp^T@V (64 tiles, 8 per wave); S = Afin*(S+U) ---
#pragma unroll
    for (int u = 0; u < 8; ++u) {
      int tt = w * 8 + u, mt = tt >> 2, nt = tt & 3;
      v8f ua = {};
#pragma unroll
      for (int kk = 0; kk < 16; kk += 4) {
        v2f a;
        a.x = Kp[(kk + 2 * h2) * 257 + mt * 16 + l15];
        a.y = Kp[(kk + 2 * h2 + 1) * 257 + mt * 16 + l15];
        v2f bb;
        bb.x = rv[(kk + 2 * h2) * 72 + nt * 16 + l15];
        bb.y = rv[(kk + 2 * h2 + 1) * 72 + nt * 16 + l15];
        ua = wmma4(a, bb, ua);
      }
#pragma unroll
      for (int r = 0; r < 8; ++r) {
        int i = mt * 16 + r + 8 * h2, j = nt * 16 + l15;
        float af = Afin[i & 127];
        S[i * 66 + j] = af * (S[i * 66 + j] + ua[r]);
      }
    }
    // --- D: O += tril(P)@V (waves 0-3), store chunk output ---
    if (w < 4) {
#pragma unroll
      for (int kk = 0; kk < 16; kk += 4) {
        v2f a;
        a.x = Pm[l15 * 20 + kk + 2 * h2];
        a.y = Pm[l15 * 20 + kk + 2 * h2 + 1];
        v2f bb;
        bb.x = rv[(kk + 2 * h2) * 72 + w * 16 + l15];
        bb.y = rv[(kk + 2 * h2 + 1) * 72 + w * 16 + l15];
        acc = wmma4(a, bb, acc);
      }
#pragma unroll
      for (int r = 0; r < 8; ++r) {
        int t = r + 8 * h2;
        Oat[baseBH + (size_t)(t0 + t) * EMBED + jb * 64 + w * 16 + l15] = acc[r];
      }
    }
    wait_async();        // prefetched chunk nc+1 landed
    __syncthreads();
  }
}

// ---------------------------------------------------------------------------
// out = LayerNorm(o * gate) * norm_w  (in-place on Oat; one block per row)
// ---------------------------------------------------------------------------
__global__ __launch_bounds__(256)
void ln_gate(float* __restrict__ Oat, const float* __restrict__ Gate,
             const float* __restrict__ Nw) {
  __shared__ float r1[256], r2[256];
  int row = blockIdx.x;
  size_t base = (size_t)row * EMBED;
  float vals[4];
  float s = 0.f, sq = 0.f;
#pragma unroll
  for (int i = 0; i < 4; ++i) {
    int c = threadIdx.x + i * 256;
    float v = Oat[base + c] * Gate[base + c];
    vals[i] = v; s += v; sq += v * v;
  }
  r1[threadIdx.x] = s; r2[threadIdx.x] = sq;
  __syncthreads();
  for (int off = 128; off > 0; off >>= 1) {
    if (threadIdx.x < off) {
      r1[threadIdx.x] += r1[threadIdx.x + off];
      r2[threadIdx.x] += r2[threadIdx.x + off];
    }
    __syncthreads();
  }
  float mu = r1[0] / (float)EMBED;
  float var = r2[0] / (float)EMBED - mu * mu;
  float rs = rsqrtf(var + 1e-5f);
#pragma unroll
  for (int i = 0; i < 4; ++i) {
    int c = threadIdx.x + i * 256;
    Oat[base + c] = (vals[i] - mu) * rs * Nw[c];
  }
}

// ---------------------------------------------------------------------------
extern "C" void kernel_launch(void* const* d_in, const int* in_sizes, int n_in,
                              void* d_out, int out_size, void* d_ws,
                              size_t ws_size, hipStream_t stream) {
  const float* x   = (const float*)d_in[0];
  const float* wq  = (const float*)d_in[1];
  const float* wk  = (const float*)d_in[2];
  const float* wv  = (const float*)d_in[3];
  const float* wo  = (const float*)d_in[4];
  const float* wg1 = (const float*)d_in[5];
  const float* wg2 = (const float*)d_in[6];
  const float* nw  = (const float*)d_in[7];
  float* out = (float*)d_out;
  float* ws  = (float*)d_ws;
  const size_t NE = (size_t)BT * EMBED;
  float* q    = ws;
  float* k    = ws + NE;
  float* v    = ws + 2 * NE;
  float* cosb = ws + 3 * NE;
  float* sinb = ws + 4 * NE;
  float* oat  = ws + 5 * NE;
  float* gate = cosb;  // cos dead after scan
  float* t1   = sinb;  // sin dead after scan

  dim3 blk(256);
  dim3 gEE(BT / 128, EMBED / 64);
  hipLaunchKernelGGL(gemm_wmma_f32, gEE, blk, 0, stream, x, wq, q, BT, EMBED, EMBED, 1);
  hipLaunchKernelGGL(gemm_wmma_f32, gEE, blk, 0, stream, x, wk, k, BT, EMBED, EMBED, 2);
  hipLaunchKernelGGL(gemm_wmma_f32, gEE, blk, 0, stream, x, wv, v, BT, EMBED, EMBED, 0);
  hipLaunchKernelGGL(cumsum_trig, dim3((BATCH * EMBED) / 256), blk, 0, stream, k, cosb, sinb);
  hipLaunchKernelGGL(hgru3_scan, dim3(2, NHEAD, BATCH), blk, 0, stream, q, k, v, cosb, sinb, oat);
  hipLaunchKernelGGL(gemm_wmma_f32, dim3(BT / 128, DHEAD / 64), blk, 0, stream, x, wg1, t1, BT, DHEAD, EMBED, 0);
  hipLaunchKernelGGL(gemm_wmma_f32, gEE, blk, 0, stream, t1, wg2, gate, BT, EMBED, DHEAD, 2);
  hipLaunchKernelGGL(ln_gate, dim3(BT), blk, 0, stream, oat, gate, nw);
  hipLaunchKernelGGL(gemm_wmma_f32, gEE, blk, 0, stream, oat, wo, out, BT, EMBED, EMBED, 0);
}